// SelectiveScan_70600672411796
// MI455X (gfx1250) — compile-verified
//
#include <hip/hip_runtime.h>
#include <math.h>

// ---------------------------------------------------------------------------
// SelectiveScan (Mamba-style S4 scan) for MI455X / gfx1250, wave32 + WMMA f32.
//   Bx    = x @ B_in                      (WMMA f32 16x16x4, B_in staged in LDS)
//   decay = exp(softplus(Bx) * -exp(A))
//   h_t   = h_{t-1} * decay_t + Bx_t      (chunked parallel scan, 64ch x 4096)
//   y     = hs @ C + x * D                (WMMA f32 16x16x4, K=16, fused epi)
// ---------------------------------------------------------------------------

typedef __attribute__((ext_vector_type(2))) float v2f;
typedef __attribute__((ext_vector_type(4))) float v4f;
typedef __attribute__((ext_vector_type(8))) float v8f;

#define D_MODEL 1024
#define D_STATE 16
#define BATCH 4
#define SEQLEN 4096
#define NROWS (BATCH * SEQLEN)      // 16384 rows of x / y
#define CHUNK 64                    // scan chunk length
#define NCHUNK (SEQLEN / CHUNK)     // 64 chunks
#define NCHAN (BATCH * D_STATE)     // 64 independent scan channels

// ---------------------------------------------------------------------------
// Phase 1: Bx = x @ B_in ; decay = exp(softplus(Bx) * -exp(A))
// One wave per 16-row tile of x; 4 waves / block. Full B_in (64 KB) staged in
// LDS once per block (WGP has 320 KB), WMMA B-fragments come from ds_load.
// Two accumulators break the WMMA accumulate chain.
// ---------------------------------------------------------------------------
__global__ __launch_bounds__(128) void ss_gemm1_kernel(
    const float* __restrict__ x,      // [NROWS, D_MODEL]
    const float* __restrict__ A,      // [D_STATE]
    const float* __restrict__ Bin,    // [D_MODEL, D_STATE]
    float* __restrict__ bx_ws,        // [NROWS, D_STATE]
    float* __restrict__ decay_ws)     // [NROWS, D_STATE]
{
    __shared__ float ldsB[D_MODEL * D_STATE];   // 64 KB

    // Cooperative, coalesced b128 staging of B_in into LDS.
    {
        const v4f* __restrict__ src = (const v4f*)Bin;
        v4f* dst = (v4f*)ldsB;
        for (int i = threadIdx.x; i < (D_MODEL * D_STATE) / 4; i += 128)
            dst[i] = src[i];
    }
    __syncthreads();

    const int wave  = threadIdx.x >> 5;
    const int lane  = threadIdx.x & 31;
    const int m     = lane & 15;      // row-in-tile (A) / state column (B,C)
    const int half  = lane >> 4;      // K-pair selector
    const int rtile = blockIdx.x * 4 + wave;           // 1024 tiles total
    const int rbase = rtile * 16;

    const float aneg = -expf(A[m]);   // per-state -exp(A[s]), s == m

    const float* __restrict__ xrow = x + (size_t)(rbase + m) * D_MODEL;

    v8f acc0 = {};
    v8f acc1 = {};

    for (int k0 = 0; k0 < D_MODEL; k0 += 8) {
        // --- WMMA #0 : K = k0..k0+3 ---
        v2f a0 = *(const v2f*)(xrow + k0 + 2 * half);
        v2f b0;
        b0.x = ldsB[(k0 + 2 * half + 0) * D_STATE + m];
        b0.y = ldsB[(k0 + 2 * half + 1) * D_STATE + m];
        acc0 = __builtin_amdgcn_wmma_f32_16x16x4_f32(
            false, a0, false, b0, (short)0, acc0, false, false);

        // --- WMMA #1 : K = k0+4..k0+7 (independent accumulator) ---
        v2f a1 = *(const v2f*)(xrow + k0 + 4 + 2 * half);
        v2f b1;
        b1.x = ldsB[(k0 + 4 + 2 * half + 0) * D_STATE + m];
        b1.y = ldsB[(k0 + 4 + 2 * half + 1) * D_STATE + m];
        acc1 = __builtin_amdgcn_wmma_f32_16x16x4_f32(
            false, a1, false, b1, (short)0, acc1, false, false);
    }

    // C/D layout: VGPR v, lane -> element (M = v + 8*half, N = m)
    #pragma unroll
    for (int v = 0; v < 8; ++v) {
        float bx  = acc0[v] + acc1[v];
        int   r   = rbase + v + 8 * half;
        float sp  = (bx > 20.0f) ? bx : log1pf(expf(bx));   // softplus
        float dec = expf(sp * aneg);
        bx_ws[r * D_STATE + m]    = bx;
        decay_ws[r * D_STATE + m] = dec;
    }
}

// ---------------------------------------------------------------------------
// Phase 2a: per-(batch, chunk) partial scan with h0 = 0, all 16 states per
// thread held in v4f registers. Every timestep is 2x (4 contiguous b128 loads).
// Produces chunk decay product P and chunk partial state H, layout [NCHUNK][NCHAN].
// ---------------------------------------------------------------------------
__global__ void ss_scan_partial_kernel(
    const float* __restrict__ bx_ws,
    const float* __restrict__ decay_ws,
    float* __restrict__ P,            // [NCHUNK, NCHAN]
    float* __restrict__ H)            // [NCHUNK, NCHAN]
{
    int t = blockIdx.x * blockDim.x + threadIdx.x;   // 0 .. BATCH*NCHUNK-1
    if (t >= BATCH * NCHUNK) return;
    int b = t >> 6;                    // batch
    int c = t & (NCHUNK - 1);          // chunk

    size_t base = ((size_t)b * SEQLEN + (size_t)c * CHUNK) * D_STATE;
    const v4f one = {1.0f, 1.0f, 1.0f, 1.0f};
    v4f h[4] = {};
    v4f p[4] = {one, one, one, one};

    for (int j = 0; j < CHUNK; ++j) {
        const v4f* dv = (const v4f*)(decay_ws + base + (size_t)j * D_STATE);
        const v4f* bv = (const v4f*)(bx_ws    + base + (size_t)j * D_STATE);
        #pragma unroll
        for (int q = 0; q < 4; ++q) {
            v4f d = dv[q], v = bv[q];
            h[q] = h[q] * d + v;
            p[q] = p[q] * d;
        }
    }

    v4f* Pp = (v4f*)(P + c * NCHAN + b * D_STATE);
    v4f* Hp = (v4f*)(H + c * NCHAN + b * D_STATE);
    #pragma unroll
    for (int q = 0; q < 4; ++q) { Pp[q] = p[q]; Hp[q] = h[q]; }
}

// ---------------------------------------------------------------------------
// Phase 2b: sequential scan over chunk carries (64 channels x 64 chunks).
// Single tiny block; critical path is only NCHUNK steps.
// ---------------------------------------------------------------------------
__global__ void ss_scan_carry_kernel(
    const float* __restrict__ P,
    const float* __restrict__ H,
    float* __restrict__ carry)        // [NCHUNK, NCHAN] carry entering chunk c
{
    int ch = threadIdx.x;             // 64 threads, one per channel
    float cy = 0.0f;
    for (int c = 0; c < NCHUNK; ++c) {
        carry[c * NCHAN + ch] = cy;
        cy = cy * P[c * NCHAN + ch] + H[c * NCHAN + ch];
    }
}

// ---------------------------------------------------------------------------
// Phase 2c: replay each chunk with its true carry, emit hs (b128 stores).
// ---------------------------------------------------------------------------
__global__ void ss_scan_final_kernel(
    const float* __restrict__ bx_ws,
    const float* __restrict__ decay_ws,
    const float* __restrict__ carry,
    float* __restrict__ hs)           // [NROWS, D_STATE]
{
    int t = blockIdx.x * blockDim.x + threadIdx.x;
    if (t >= BATCH * NCHUNK) return;
    int b = t >> 6;
    int c = t & (NCHUNK - 1);

    size_t base = ((size_t)b * SEQLEN + (size_t)c * CHUNK) * D_STATE;
    const v4f* cv = (const v4f*)(carry + c * NCHAN + b * D_STATE);
    v4f h[4];
    #pragma unroll
    for (int q = 0; q < 4; ++q) h[q] = cv[q];

    for (int j = 0; j < CHUNK; ++j) {
        const v4f* dv = (const v4f*)(decay_ws + base + (size_t)j * D_STATE);
        const v4f* bv = (const v4f*)(bx_ws    + base + (size_t)j * D_STATE);
        v4f*       ov = (v4f*)(hs + base + (size_t)j * D_STATE);
        #pragma unroll
        for (int q = 0; q < 4; ++q) {
            v4f d = dv[q], v = bv[q];
            h[q] = h[q] * d + v;
            ov[q] = h[q];
        }
    }
}

// ---------------------------------------------------------------------------
// Phase 3: y = hs @ C + x * D. One wave per 16x16 output tile; K=16 via four
// WMMA f32 16x16x4 (dual accumulators), fused x*D epilogue.
// ---------------------------------------------------------------------------
__global__ __launch_bounds__(128) void ss_gemm2_kernel(
    const float* __restrict__ hs,     // [NROWS, D_STATE]
    const float* __restrict__ Cm,     // [D_STATE, D_MODEL]
    const float* __restrict__ x,      // [NROWS, D_MODEL]
    const float* __restrict__ Dv,     // [D_MODEL]
    float* __restrict__ y)            // [NROWS, D_MODEL]
{
    const int wave  = threadIdx.x >> 5;
    const int lane  = threadIdx.x & 31;
    const int m     = lane & 15;
    const int half  = lane >> 4;
    const int tile  = blockIdx.x * 4 + wave;   // 65536 tiles
    const int rtile = tile >> 6;               // 1024 row tiles
    const int ctile = tile & 63;               // 64 col tiles
    const int r     = rtile * 16 + m;
    const int e     = ctile * 16 + m;

    // Warm the x tile (lowers to global_prefetch_b8 on gfx1250).
    __builtin_prefetch(x + (size_t)r * D_MODEL + ctile * 16, 0, 1);

    v8f acc0 = {};
    v8f acc1 = {};

    // k0 = 0 and 8 -> acc0 ; k0 = 4 and 12 -> acc1
    #pragma unroll
    for (int kk = 0; kk < 2; ++kk) {
        int k0 = kk * 8;
        v2f a0 = *(const v2f*)(hs + (size_t)r * D_STATE + k0 + 2 * half);
        v2f b0;
        b0.x = Cm[(k0 + 2 * half + 0) * D_MODEL + e];
        b0.y = Cm[(k0 + 2 * half + 1) * D_MODEL + e];
        acc0 = __builtin_amdgcn_wmma_f32_16x16x4_f32(
            false, a0, false, b0, (short)0, acc0, false, false);

        int k1 = k0 + 4;
        v2f a1 = *(const v2f*)(hs + (size_t)r * D_STATE + k1 + 2 * half);
        v2f b1;
        b1.x = Cm[(k1 + 2 * half + 0) * D_MODEL + e];
        b1.y = Cm[(k1 + 2 * half + 1) * D_MODEL + e];
        acc1 = __builtin_amdgcn_wmma_f32_16x16x4_f32(
            false, a1, false, b1, (short)0, acc1, false, false);
    }

    const float dcoef = Dv[e];
    #pragma unroll
    for (int v = 0; v < 8; ++v) {
        int ro = rtile * 16 + v + 8 * half;
        size_t idx = (size_t)ro * D_MODEL + e;
        y[idx] = (acc0[v] + acc1[v]) + x[idx] * dcoef;
    }
}

// ---------------------------------------------------------------------------
// Host-side launcher (graph-capture safe: stream-only, no sync/alloc).
// ---------------------------------------------------------------------------
extern "C" void kernel_launch(void* const* d_in, const int* in_sizes, int n_in,
                              void* d_out, int out_size, void* d_ws, size_t ws_size,
                              hipStream_t stream) {
    const float* x   = (const float*)d_in[0];   // [4, 4096, 1024]
    const float* A   = (const float*)d_in[1];   // [1, 16]
    const float* Bin = (const float*)d_in[2];   // [1024, 16]
    const float* Cm  = (const float*)d_in[3];   // [16, 1024]
    const float* Dv  = (const float*)d_in[4];   // [1024]
    float*       y   = (float*)d_out;           // [4, 4096, 1024]

    // Workspace layout (floats)
    float* ws       = (float*)d_ws;
    float* bx_ws    = ws;                                   // 262144
    float* decay_ws = bx_ws + (size_t)NROWS * D_STATE;      // 262144
    float* hs_ws    = decay_ws + (size_t)NROWS * D_STATE;   // 262144
    float* P_ws     = hs_ws + (size_t)NROWS * D_STATE;      // 4096
    float* H_ws     = P_ws + NCHAN * NCHUNK;                // 4096
    float* carry_ws = H_ws + NCHAN * NCHUNK;                // 4096

    // Phase 1: Bx + decay (1024 row tiles, 4 waves/block, B_in in LDS)
    ss_gemm1_kernel<<<256, 128, 0, stream>>>(x, A, Bin, bx_ws, decay_ws);

    // Phase 2: chunked parallel scan (256 tasks, 16 states per thread in v4f)
    ss_scan_partial_kernel<<<4, 64, 0, stream>>>(bx_ws, decay_ws, P_ws, H_ws);
    ss_scan_carry_kernel<<<1, NCHAN, 0, stream>>>(P_ws, H_ws, carry_ws);
    ss_scan_final_kernel<<<4, 64, 0, stream>>>(bx_ws, decay_ws, carry_ws, hs_ws);

    // Phase 3: y = hs @ C + x * D (65536 tiles, 4 waves/block)
    ss_gemm2_kernel<<<16384, 128, 0, stream>>>(hs_ws, Cm, x, Dv, y);
}